// STVQVae_85169201480001
// MI455X (gfx1250) — compile-verified
//
#include <hip/hip_runtime.h>

// VQ-VAE encoder quantization for gfx1250 (MI455X).
// z_e: [B=8, D=64, H=64, W=64] fp32, codebook: [K=8192, D=64] fp32.
// out: z_q = codebook[argmin_k ||x - c_k||^2] scattered back to NCHW.
// Core compute: fp32 WMMA (V_WMMA_F32_16X16X4_F32). Each wave owns TWO 16-row
// x-tiles (32 rows) and runs two independent accumulator chains per B-chunk
// loaded from LDS: doubles arithmetic intensity vs LDS and hides WMMA RAW
// latency via interleaved independent chains.

typedef __attribute__((ext_vector_type(2))) float v2f;
typedef __attribute__((ext_vector_type(8))) float v8f;

#define DDIM 64
#define HW   4096            // H*W
#define TK   128             // codebook rows staged in LDS per iteration
#define CB_STRIDE 68         // padded LDS row stride (floats) -> conflict-free b64 reads
#define WAVES_PER_BLOCK 8
#define BLOCK_THREADS 256
#define ROWS_PER_WAVE 32     // two 16-row x-tiles per wave

// ---- kernel 1: per-code squared norms --------------------------------------
__global__ void __launch_bounds__(256)
cnorm_kernel(const float* __restrict__ cb, float* __restrict__ cn, int K) {
    int i = blockIdx.x * blockDim.x + threadIdx.x;
    if (i >= K) return;
    const float4* row = (const float4*)(cb + (size_t)i * DDIM);
    float s = 0.f;
#pragma unroll
    for (int j = 0; j < DDIM / 4; ++j) {
        float4 v = row[j];
        s += v.x * v.x + v.y * v.y + v.z * v.z + v.w * v.w;
    }
    cn[i] = s;
}

// ---- kernel 2: WMMA argmin + gather-write ----------------------------------
__global__ void __launch_bounds__(BLOCK_THREADS)
vq_argmin_kernel(const float* __restrict__ z_e,       // [B, D, HW]
                 const float* __restrict__ codebook,  // [K, D]
                 const float* __restrict__ cnorm,     // [K]
                 float* __restrict__ z_q,             // [B, D, HW]
                 int K) {
    __shared__ __align__(16) float s_cb[TK * CB_STRIDE];
    __shared__ float s_cn[TK];
    __shared__ int   s_idx[WAVES_PER_BLOCK * ROWS_PER_WAVE];

    const int tid  = threadIdx.x;
    const int wave = tid >> 5;
    const int lane = tid & 31;
    const int r    = lane & 15;            // A-row / B-col within 16-tile
    const int dlt  = (lane >> 4) << 1;     // k sub-offset: 0 (lanes 0-15) / 2 (16-31)

    // This wave's 32 consecutive flattened (b,h,w) rows.
    const int task = blockIdx.x * WAVES_PER_BLOCK + wave;
    const int m0   = task * ROWS_PER_WAVE;
    const int bb   = m0 / HW;              // 32 | HW, so one batch per task
    const int hw0  = m0 - bb * HW;
    const float* zb = z_e + (size_t)bb * DDIM * HW + hw0;

    // A operand register cache for both 16-row tiles:
    // lane holds row r (tile0) / row r+16 (tile1), elems {4c+dlt, 4c+dlt+1}.
    v2f A0[16], A1[16];
#pragma unroll
    for (int c = 0; c < 16; ++c) {
        int k = 4 * c + dlt;
        A0[c].x = zb[(size_t)k * HW + r];
        A0[c].y = zb[(size_t)(k + 1) * HW + r];
        A1[c].x = zb[(size_t)k * HW + 16 + r];
        A1[c].y = zb[(size_t)(k + 1) * HW + 16 + r];
    }

    float best0[8], best1[8];
    int   bidx0[8], bidx1[8];
#pragma unroll
    for (int v = 0; v < 8; ++v) {
        best0[v] = 3.4e38f; bidx0[v] = 0;
        best1[v] = 3.4e38f; bidx1[v] = 0;
    }

    const int numTiles = K / TK;
    for (int ct = 0; ct < numTiles; ++ct) {
        __syncthreads();
        // Cooperative stage: TK codebook rows -> LDS (padded), + norms.
        {
            const float4* src = (const float4*)(codebook + (size_t)ct * TK * DDIM);
#pragma unroll
            for (int i = 0; i < (TK * DDIM / 4) / BLOCK_THREADS; ++i) {
                int flat = tid + i * BLOCK_THREADS;
                int row  = flat >> 4;          // 16 float4 per 64-float row
                int c4   = flat & 15;
                float4 val = src[row * 16 + c4];
                *(float4*)&s_cb[row * CB_STRIDE + (c4 << 2)] = val;
            }
            if (tid < TK) s_cn[tid] = cnorm[ct * TK + tid];
        }
        __syncthreads();

#pragma unroll 1
        for (int sub = 0; sub < TK / 16; ++sub) {
            const int nloc = (sub << 4) + r;
            const float* brow = &s_cb[nloc * CB_STRIDE + dlt];
            v8f acc0 = {};
            v8f acc1 = {};
#pragma unroll
            for (int c = 0; c < 16; ++c) {
                v2f bv = *(const v2f*)(brow + 4 * c);
                // Two independent chains off the same B chunk.
                acc0 = __builtin_amdgcn_wmma_f32_16x16x4_f32(
                    false, A0[c], false, bv, (short)0, acc0, false, false);
                acc1 = __builtin_amdgcn_wmma_f32_16x16x4_f32(
                    false, A1[c], false, bv, (short)0, acc1, false, false);
            }
            const float cn = s_cn[nloc];
            const int   n  = ct * TK + nloc;
#pragma unroll
            for (int v = 0; v < 8; ++v) {
                float d20 = fmaf(-2.0f, acc0[v], cn);   // ||c||^2 - 2 x.c
                float d21 = fmaf(-2.0f, acc1[v], cn);
                if (d20 < best0[v]) { best0[v] = d20; bidx0[v] = n; }
                if (d21 < best1[v]) { best1[v] = d21; bidx1[v] = n; }
            }
        }
    }

    // Min+index reduction across the 16 lanes of each half-wave
    // (xor masks 1,2,4,8 never cross the 16-lane boundary in wave32).
#pragma unroll
    for (int v = 0; v < 8; ++v) {
        float bv0 = best0[v]; int bi0 = bidx0[v];
        float bv1 = best1[v]; int bi1 = bidx1[v];
#pragma unroll
        for (int off = 1; off <= 8; off <<= 1) {
            float ov0 = __shfl_xor(bv0, off, 32);
            int   oi0 = __shfl_xor(bi0, off, 32);
            float ov1 = __shfl_xor(bv1, off, 32);
            int   oi1 = __shfl_xor(bi1, off, 32);
            if (ov0 < bv0 || (ov0 == bv0 && oi0 < bi0)) { bv0 = ov0; bi0 = oi0; }
            if (ov1 < bv1 || (ov1 == bv1 && oi1 < bi1)) { bv1 = ov1; bi1 = oi1; }
        }
        if (r == 0) {
            int slot = v + ((lane >> 4) << 3);      // row within 16-tile
            s_idx[wave * ROWS_PER_WAVE + slot]      = bi0;
            s_idx[wave * ROWS_PER_WAVE + 16 + slot] = bi1;
        }
    }
    __syncthreads();

    // Scatter selected code vectors back to NCHW:
    // out[bb, d, hw0 + row] = codebook[idx[row]*64 + d], rows 0..31
    float* ob = z_q + (size_t)bb * DDIM * HW + hw0;
    const int myIdx0 = s_idx[wave * ROWS_PER_WAVE + r];
    const int myIdx1 = s_idx[wave * ROWS_PER_WAVE + 16 + r];
    const int dbase = lane >> 4;
#pragma unroll
    for (int j = 0; j < 32; ++j) {
        int d = (j << 1) + dbase;
        ob[(size_t)d * HW + r]      = codebook[(size_t)myIdx0 * DDIM + d];
        ob[(size_t)d * HW + 16 + r] = codebook[(size_t)myIdx1 * DDIM + d];
    }
}

// ---- launch ----------------------------------------------------------------
extern "C" void kernel_launch(void* const* d_in, const int* in_sizes, int n_in,
                              void* d_out, int out_size, void* d_ws, size_t ws_size,
                              hipStream_t stream) {
    const float* z_e = (const float*)d_in[0];
    const float* cb  = (const float*)d_in[1];
    float* out       = (float*)d_out;

    const int N = in_sizes[0] / DDIM;   // 32768 flattened spatial positions
    const int K = in_sizes[1] / DDIM;   // 8192 codes

    float* cnorm = (float*)d_ws;        // K floats of scratch

    cnorm_kernel<<<(K + 255) / 256, 256, 0, stream>>>(cb, cnorm, K);

    const int tasks = N / ROWS_PER_WAVE;  // 1024 wave-tasks, 8 per block
    vq_argmin_kernel<<<tasks / WAVES_PER_BLOCK, BLOCK_THREADS, 0, stream>>>(
        z_e, cb, cnorm, out, K);
}